// PretrainedBidafModelUtils_14731737825591
// MI455X (gfx1250) — compile-verified
//
#include <hip/hip_runtime.h>
#include <math.h>

typedef __attribute__((ext_vector_type(2))) float v2f;
typedef __attribute__((ext_vector_type(8))) float v8f;

#define BB 16
#define CC 4096
#define QQ 128
#define DD 256
#define CT 64
#define KC 64
#define NEG_VAL (-10000000.0f)

// ---------------------------------------------------------------------------
// K1: px[b,c] = dot(P[b,c,:], w_x) ; qy[b,q] = dot(Q[b,q,:], w_y)
// one wave (32 lanes) per row; 8 strided loads + shuffle reduction
// ---------------------------------------------------------------------------
__global__ void k1_dots(const float* __restrict__ q, const float* __restrict__ p,
                        const float* __restrict__ wx, const float* __restrict__ wy,
                        float* __restrict__ px, float* __restrict__ qy)
{
    const int wv = threadIdx.x >> 5;
    const int lane = threadIdx.x & 31;
    const int row = blockIdx.x * 8 + wv;

    const float* src;
    const float* wgt;
    float* dst;
    int di;
    if (row < BB * CC) {
        src = p + (size_t)row * DD; wgt = wx; dst = px; di = row;
    } else {
        const int r2 = row - BB * CC;
        src = q + (size_t)r2 * DD; wgt = wy; dst = qy; di = r2;
    }
    float v = 0.0f;
#pragma unroll
    for (int i = 0; i < 8; ++i) {
        const int k = lane + i * 32;
        v += src[k] * wgt[k];
    }
#pragma unroll
    for (int off = 16; off >= 1; off >>= 1)
        v += __shfl_xor(v, off, 32);
    if (lane == 0) dst[di] = v;
}

// ---------------------------------------------------------------------------
// K2: per (b, 64-row c-tile):
//   sim = (P .* w_xy) @ Q^T + px + qy + bias      (WMMA f32 16x16x4)
//   att = masked_softmax(sim, qmask) over q; qp_sim = rowmax of masked sim
//   pq  = att @ Q                                  (WMMA f32 16x16x4)
//   out[...,256:512] = pq ; out[...,512:768] = P .* pq
// 256 threads = 8 waves; wave w: M-tile = w&3, N-range = (w>>2)
// ---------------------------------------------------------------------------
__global__ void __launch_bounds__(256)
k2_attention(const float* __restrict__ q, const float* __restrict__ p,
             const float* __restrict__ qmask, const float* __restrict__ wxy,
             const float* __restrict__ bias,
             const float* __restrict__ px, const float* __restrict__ qy,
             float* __restrict__ qp_sim, float* __restrict__ out)
{
    // smem union: [ sA 64x68 | sQ 128x68 ]  reused as  [ sS 64x132 ]
    __shared__ float smem[64 * 68 + 128 * 68];   // 52.2 KB
    __shared__ float redA[256];
    __shared__ float redB[256];
    float* sA = smem;
    float* sQ = smem + 64 * 68;
    float* sS = smem;

    const int tid = threadIdx.x;
    const int lane = tid & 31;
    const int w = tid >> 5;
    const int lo = lane & 15;
    const int hi = lane >> 4;
    const int b = blockIdx.y;
    const int c0 = blockIdx.x * CT;
    const int mt = w & 3;
    const int ntb = (w >> 2) * 4;

    const float* Pb = p + (size_t)(b * CC) * DD;
    const float* Qb = q + (size_t)(b * QQ) * DD;

    v8f acc[4] = {};

    // ---- GEMM1: sim = (P .* w_xy) @ Q^T, K-chunked through LDS ----
    for (int ch = 0; ch < DD / KC; ++ch) {
        const int kk0 = ch * KC;
        // stage A chunk (64 x 64), fold in w_xy
#pragma unroll
        for (int i = 0; i < 4; ++i) {
            const int idx = tid + i * 256;
            const int row = idx >> 4;
            const int c4 = (idx & 15) * 4;
            const float4 pv = *(const float4*)(Pb + (c0 + row) * DD + kk0 + c4);
            const float4 wv = *(const float4*)(wxy + kk0 + c4);
            float* dst = sA + row * 68 + c4;
            dst[0] = pv.x * wv.x; dst[1] = pv.y * wv.y;
            dst[2] = pv.z * wv.z; dst[3] = pv.w * wv.w;
        }
        // stage Q chunk (128 x 64)
#pragma unroll
        for (int i = 0; i < 8; ++i) {
            const int idx = tid + i * 256;
            const int row = idx >> 4;
            const int c4 = (idx & 15) * 4;
            const float4 qv = *(const float4*)(Qb + row * DD + kk0 + c4);
            float* dst = sQ + row * 68 + c4;
            dst[0] = qv.x; dst[1] = qv.y; dst[2] = qv.z; dst[3] = qv.w;
        }
        __syncthreads();

#pragma unroll
        for (int kk = 0; kk < KC; kk += 4) {
            const int ac = kk + 2 * hi;
            v2f a;
            a.x = sA[(mt * 16 + lo) * 68 + ac];
            a.y = sA[(mt * 16 + lo) * 68 + ac + 1];
#pragma unroll
            for (int j = 0; j < 4; ++j) {
                const int nt = ntb + j;
                v2f bf;
                bf.x = sQ[(nt * 16 + lo) * 68 + ac];
                bf.y = sQ[(nt * 16 + lo) * 68 + ac + 1];
                acc[j] = __builtin_amdgcn_wmma_f32_16x16x4_f32(
                    false, a, false, bf, (short)0, acc[j], false, false);
            }
        }
        __syncthreads();
    }

    // ---- write sim tile to LDS with rank-1 terms ----
    const float biasv = bias[0];
#pragma unroll
    for (int j = 0; j < 4; ++j) {
        const int nt = ntb + j;
#pragma unroll
        for (int r = 0; r < 8; ++r) {
            const int rloc = mt * 16 + r + 8 * hi;
            const int qq = nt * 16 + lo;
            const float s = acc[j][r] + px[b * CC + c0 + rloc] + qy[b * QQ + qq] + biasv;
            sS[rloc * 132 + qq] = s;
        }
    }
    __syncthreads();

    // ---- masked softmax over q (exact reference semantics) ----
    {
        const int row = tid >> 2;
        const int seg = tid & 3;
        const int qb0 = seg * 32;
        const float* qmb = qmask + b * QQ;

        float lmax = -3.4e38f, lqp = -3.4e38f;
        for (int i = 0; i < 32; ++i) {
            const int qq = qb0 + i;
            const float m = qmb[qq];
            const float y = sS[row * 132 + qq] * m;
            lmax = fmaxf(lmax, y);
            lqp = fmaxf(lqp, y + (1.0f - m) * NEG_VAL);
        }
        redA[tid] = lmax; redB[tid] = lqp;
        __syncthreads();
        const int rb = row * 4;
        const float mx = fmaxf(fmaxf(redA[rb], redA[rb + 1]), fmaxf(redA[rb + 2], redA[rb + 3]));
        if (seg == 0) {
            const float qpm = fmaxf(fmaxf(redB[rb], redB[rb + 1]), fmaxf(redB[rb + 2], redB[rb + 3]));
            qp_sim[b * CC + c0 + row] = qpm;
        }
        __syncthreads();
        float se = 0.0f, sem = 0.0f;
        for (int i = 0; i < 32; ++i) {
            const int qq = qb0 + i;
            const float m = qmb[qq];
            const float e = __expf(sS[row * 132 + qq] * m - mx);
            se += e; sem += e * m;
        }
        redA[tid] = se; redB[tid] = sem;
        __syncthreads();
        const float Se = redA[rb] + redA[rb + 1] + redA[rb + 2] + redA[rb + 3];
        const float Sem = redB[rb] + redB[rb + 1] + redB[rb + 2] + redB[rb + 3];
        const float inv = 1.0f / (Sem + 1e-13f * Se);
        __syncthreads();
        for (int i = 0; i < 32; ++i) {
            const int qq = qb0 + i;
            const float m = qmb[qq];
            const float x = sS[row * 132 + qq];
            sS[row * 132 + qq] = m * __expf(x * m - mx) * inv;   // att
        }
    }
    __syncthreads();

    // ---- GEMM2: pq = att @ Q ; fused epilogue (quarters 1 & 2) ----
    for (int nt8 = 0; nt8 < 8; ++nt8) {
        const int nt = (w >> 2) * 8 + nt8;
        const int n0 = nt * 16;
        v8f acc2 = {};
#pragma unroll
        for (int kk = 0; kk < QQ; kk += 4) {
            const int kc = kk + 2 * hi;
            v2f a;
            a.x = sS[(mt * 16 + lo) * 132 + kc];
            a.y = sS[(mt * 16 + lo) * 132 + kc + 1];
            v2f bf;
            bf.x = Qb[kc * DD + n0 + lo];
            bf.y = Qb[(kc + 1) * DD + n0 + lo];
            acc2 = __builtin_amdgcn_wmma_f32_16x16x4_f32(
                false, a, false, bf, (short)0, acc2, false, false);
        }
#pragma unroll
        for (int r = 0; r < 8; ++r) {
            const int rloc = mt * 16 + r + 8 * hi;
            const int crow = c0 + rloc;
            const int d = n0 + lo;
            const float v = acc2[r];
            const float pvv = Pb[crow * DD + d];
            const size_t ob = ((size_t)(b * CC) + crow) * 1024;
            out[ob + 256 + d] = v;
            out[ob + 512 + d] = pvv * v;
        }
    }
}

// ---------------------------------------------------------------------------
// K3a: per batch: masked-softmax stats over c of qp_sim -> {max, 1/denom}
// ---------------------------------------------------------------------------
__global__ void __launch_bounds__(256)
k3a_stats(const float* __restrict__ pmask, const float* __restrict__ qp_sim,
          float* __restrict__ stats)
{
    __shared__ float sred[256];
    const int b = blockIdx.x;
    const int t = threadIdx.x;
    const float* qs = qp_sim + b * CC;
    const float* pm = pmask + b * CC;

    float lm = -3.4e38f;
    for (int c = t; c < CC; c += 256) lm = fmaxf(lm, qs[c] * pm[c]);
    sred[t] = lm; __syncthreads();
    for (int s = 128; s > 0; s >>= 1) {
        if (t < s) sred[t] = fmaxf(sred[t], sred[t + s]);
        __syncthreads();
    }
    const float mx = sred[0];
    __syncthreads();

    float se = 0.0f, sem = 0.0f;
    for (int c = t; c < CC; c += 256) {
        const float m = pm[c];
        const float e = __expf(qs[c] * m - mx);
        se += e; sem += e * m;
    }
    sred[t] = se; __syncthreads();
    for (int s = 128; s > 0; s >>= 1) { if (t < s) sred[t] += sred[t + s]; __syncthreads(); }
    const float Se = sred[0];
    __syncthreads();
    sred[t] = sem; __syncthreads();
    for (int s = 128; s > 0; s >>= 1) { if (t < s) sred[t] += sred[t + s]; __syncthreads(); }
    const float Sem = sred[0];

    if (t == 0) {
        stats[b * 2 + 0] = mx;
        stats[b * 2 + 1] = 1.0f / (Sem + 1e-13f * Se);
    }
}

// ---------------------------------------------------------------------------
// K3b: grid (16 c-chunks, 16 batches): partial qp_vec over 256 passage rows
// thread t owns output column d = t; no atomics -> bitwise deterministic
// ---------------------------------------------------------------------------
__global__ void __launch_bounds__(256)
k3b_partial(const float* __restrict__ p, const float* __restrict__ pmask,
            const float* __restrict__ qp_sim, const float* __restrict__ stats,
            float* __restrict__ partial)
{
    __shared__ float satt[256];
    const int chunk = blockIdx.x;
    const int b = blockIdx.y;
    const int t = threadIdx.x;
    const int c0 = chunk * 256;
    const float mx = stats[b * 2 + 0];
    const float inv = stats[b * 2 + 1];

    {
        const int c = c0 + t;
        const float m = pmask[b * CC + c];
        satt[t] = m * __expf(qp_sim[b * CC + c] * m - mx) * inv;
    }
    __syncthreads();

    float acc = 0.0f;
    const float* Pb = p + (size_t)(b * CC) * DD;
#pragma unroll 8
    for (int cc = 0; cc < 256; ++cc)
        acc += satt[cc] * Pb[(size_t)(c0 + cc) * DD + t];

    partial[((size_t)(b * 16) + chunk) * DD + t] = acc;
}

// ---------------------------------------------------------------------------
// K3c: reduce 16 chunk-partials -> qp_vec[b, d]
// ---------------------------------------------------------------------------
__global__ void __launch_bounds__(256)
k3c_reduce(const float* __restrict__ partial, float* __restrict__ qp_vec)
{
    const int b = blockIdx.x;
    const int t = threadIdx.x;
    float acc = 0.0f;
#pragma unroll
    for (int ch = 0; ch < 16; ++ch)
        acc += partial[((size_t)(b * 16) + ch) * DD + t];
    qp_vec[b * DD + t] = acc;
}

// ---------------------------------------------------------------------------
// K4: float4-coalesced quarters 0 and 3:
//   out[...,0:256] = P ; out[...,768:1024] = P .* qp_vec
// ---------------------------------------------------------------------------
__global__ void k4_concat(const float* __restrict__ p, const float* __restrict__ qp_vec,
                          float* __restrict__ out)
{
    const int idx = blockIdx.x * 256 + threadIdx.x;   // float4 index over B*C*D/4
    const int d4 = idx & 63;
    const int row = idx >> 6;       // b*C + c
    const int b = row >> 12;        // C = 4096
    const float4 pv = ((const float4*)p)[(size_t)row * 64 + d4];
    const float4 qv = ((const float4*)qp_vec)[b * 64 + d4];
    float4* o = (float4*)out + (size_t)row * 256;
    o[d4] = pv;
    float4 r;
    r.x = pv.x * qv.x; r.y = pv.y * qv.y; r.z = pv.z * qv.z; r.w = pv.w * qv.w;
    o[192 + d4] = r;
}

// ---------------------------------------------------------------------------
extern "C" void kernel_launch(void* const* d_in, const int* in_sizes, int n_in,
                              void* d_out, int out_size, void* d_ws, size_t ws_size,
                              hipStream_t stream)
{
    (void)in_sizes; (void)n_in; (void)out_size; (void)ws_size;
    const float* q     = (const float*)d_in[0];   // (B,Q,D)
    const float* p     = (const float*)d_in[1];   // (B,C,D)
    const float* qmask = (const float*)d_in[2];   // (B,Q)
    const float* pmask = (const float*)d_in[3];   // (B,C)
    const float* wx    = (const float*)d_in[4];   // (D,)
    const float* wy    = (const float*)d_in[5];   // (D,)
    const float* wxy   = (const float*)d_in[6];   // (D,)
    const float* bias  = (const float*)d_in[7];   // ()
    float* out = (float*)d_out;

    float* ws = (float*)d_ws;
    float* px      = ws;                 // B*C        = 65536
    float* qy      = ws + 65536;         // B*Q        = 2048
    float* qp_sim  = ws + 67584;         // B*C        = 65536
    float* qp_vec  = ws + 133120;        // B*D        = 4096
    float* stats   = ws + 137216;        // B*2        = 32 (+pad)
    float* partial = ws + 137280;        // B*16*D     = 65536

    k1_dots<<<(BB * CC + BB * QQ) / 8, 256, 0, stream>>>(q, p, wx, wy, px, qy);

    dim3 g2(CC / CT, BB);
    k2_attention<<<g2, 256, 0, stream>>>(q, p, qmask, wxy, bias, px, qy, qp_sim, out);

    k3a_stats<<<BB, 256, 0, stream>>>(pmask, qp_sim, stats);

    dim3 g3b(CC / 256, BB);
    k3b_partial<<<g3b, 256, 0, stream>>>(p, pmask, qp_sim, stats, partial);

    k3c_reduce<<<BB, 256, 0, stream>>>(partial, qp_vec);

    k4_concat<<<(BB * CC * DD / 4) / 256, 256, 0, stream>>>(p, qp_vec, out);
}